// MultiTaskLoss_73237782331456
// MI455X (gfx1250) — compile-verified
//
#include <hip/hip_runtime.h>
#include <stdint.h>

#define TPB 256
#define ROWS 256               // rows per tile
// per tile: emotion 256*7*4 = 7168B = 448 x b128
//           fatigue 256*3*4 = 3072B = 192 x b128
//           ftarg/etarg 1024B each  =  64 x b128 each
// total 768 b128 chunks = 3 async instructions per thread (wave-uniform regions)

__constant__ float c_F2E[3][7] = {
    {0.05f, 0.02f, 0.03f, 0.40f, 0.05f, 0.40f, 0.05f},
    {0.05f, 0.05f, 0.05f, 0.05f, 0.30f, 0.05f, 0.45f},
    {0.10f, 0.15f, 0.20f, 0.02f, 0.35f, 0.03f, 0.15f},
};

struct Smem {
    float em[2][ROWS * 7];   // 14336 B, 16B aligned
    float fa[2][ROWS * 3];   //  6144 B
    int   ft[2][ROWS];       //  2048 B
    int   et[2][ROWS];       //  2048 B
    float td[3][7];
    float ent[3];
    float sred[TPB];
};

// Fast hardware reciprocal (v_rcp_f32, ~1 ulp) — avoids the IEEE div expansion.
__device__ __forceinline__ float fast_rcp(float x) {
#if __has_builtin(__builtin_amdgcn_rcpf)
    return __builtin_amdgcn_rcpf(x);
#else
    float r;
    asm("v_rcp_f32 %0, %1" : "=v"(r) : "v"(x));
    return r;
#endif
}

// CDNA5 async global -> LDS copy (tracked by ASYNCcnt), 16 bytes per lane.
__device__ __forceinline__ void async_b128(uint32_t lds_off, uint64_t gaddr) {
    asm volatile("global_load_async_to_lds_b128 %0, %1, off"
                 :: "v"(lds_off), "v"(gaddr) : "memory");
}
__device__ __forceinline__ void wait_async3() {
    asm volatile("s_wait_asynccnt 3" ::: "memory");
}
__device__ __forceinline__ void wait_async0() {
    asm volatile("s_wait_asynccnt 0" ::: "memory");
}

__device__ __forceinline__ float row_loss(float f0, float f1, float f2,
                                          const float* e7, int ft, int et,
                                          const Smem& sm) {
    // ---- focal CE over fatigue (3 classes) ----
    float m3 = fmaxf(f0, fmaxf(f1, f2));
    float s3 = __expf(f0 - m3) + __expf(f1 - m3) + __expf(f2 - m3);
    float lse3 = m3 + __logf(s3);
    float f_t = (ft == 0) ? f0 : ((ft == 1) ? f1 : f2);
    float ce = lse3 - f_t;
    float pt = __expf(-ce);
    float u = 1.0f - pt;
    float focal = u * u * ce;

    // ---- label-smoothed CE over emotion (7 classes) ----
    float e0 = e7[0], e1 = e7[1], e2 = e7[2], e3 = e7[3], e4 = e7[4], e5 = e7[5], e6 = e7[6];
    float m7 = fmaxf(fmaxf(fmaxf(e0, e1), fmaxf(e2, e3)), fmaxf(fmaxf(e4, e5), e6));
    float t0 = __expf(e0 - m7), t1 = __expf(e1 - m7), t2 = __expf(e2 - m7),
          t3 = __expf(e3 - m7), t4 = __expf(e4 - m7), t5 = __expf(e5 - m7),
          t6 = __expf(e6 - m7);
    float s7 = ((t0 + t1) + (t2 + t3)) + ((t4 + t5) + t6);
    float sum_e = ((e0 + e1) + (e2 + e3)) + ((e4 + e5) + e6);
    float lse7 = m7 + __logf(s7);
    float logp_t = e7[et] - lse7;
    const float off = 0.1f / 6.0f;
    float lsce = -((0.9f - off) * logp_t + off * (sum_e - 7.0f * lse7));

    // ---- KL(td || softmax(e)) with eps, entropy term precomputed ----
    float inv = fast_rcp(s7);                 // p_j = t_j * inv; only feeds log(p+1e-8)
    const float* td = sm.td[ft];
    float tt[7] = {t0, t1, t2, t3, t4, t5, t6};
    float kl = sm.ent[ft];
#pragma unroll
    for (int j = 0; j < 7; ++j)
        kl = fmaf(-td[j], __logf(fmaf(tt[j], inv, 1e-8f)), kl);

    return focal + 0.8f * lsce + 0.2f * kl;
}

__global__ void __launch_bounds__(TPB)
mt_loss_main(const float* __restrict__ fat, const float* __restrict__ emo,
             const int* __restrict__ ftg, const int* __restrict__ etg,
             int B, double* __restrict__ part) {
    __shared__ __align__(16) Smem sm;
    const int tid = threadIdx.x;
    const int ntiles = B / ROWS;

    // Precompute KL target rows (matches: td = M + 1e-8; td /= sum; ent = sum td*log td)
    if (tid < 3) {
        float v[7];
        float s = 0.0f;
#pragma unroll
        for (int j = 0; j < 7; ++j) { v[j] = c_F2E[tid][j] + 1e-8f; s += v[j]; }
        float invs = 1.0f / s;                // exact divide: once per block, match reference
        float e = 0.0f;
#pragma unroll
        for (int j = 0; j < 7; ++j) {
            float t = v[j] * invs;
            float lt = __logf(t);
            sm.td[tid][j] = t;
            e += t * lt;
        }
        sm.ent[tid] = e;
    }

    const uint32_t em_l0 = (uint32_t)(uintptr_t)(&sm.em[0][0]);
    const uint32_t em_l1 = (uint32_t)(uintptr_t)(&sm.em[1][0]);
    const uint32_t fa_l0 = (uint32_t)(uintptr_t)(&sm.fa[0][0]);
    const uint32_t fa_l1 = (uint32_t)(uintptr_t)(&sm.fa[1][0]);
    const uint32_t ft_l0 = (uint32_t)(uintptr_t)(&sm.ft[0][0]);
    const uint32_t ft_l1 = (uint32_t)(uintptr_t)(&sm.ft[1][0]);
    const uint32_t et_l0 = (uint32_t)(uintptr_t)(&sm.et[0][0]);
    const uint32_t et_l1 = (uint32_t)(uintptr_t)(&sm.et[1][0]);

    auto issue_tile = [&](int tile, int b) {
        uint64_t em_g = (uint64_t)(uintptr_t)emo + (uint64_t)tile * (ROWS * 7 * 4);
        uint64_t fa_g = (uint64_t)(uintptr_t)fat + (uint64_t)tile * (ROWS * 3 * 4);
        uint64_t ft_g = (uint64_t)(uintptr_t)ftg + (uint64_t)tile * (ROWS * 4);
        uint64_t et_g = (uint64_t)(uintptr_t)etg + (uint64_t)tile * (ROWS * 4);
        uint32_t em_l = b ? em_l1 : em_l0;
        uint32_t fa_l = b ? fa_l1 : fa_l0;
        uint32_t ft_l = b ? ft_l1 : ft_l0;
        uint32_t et_l = b ? et_l1 : et_l0;
#pragma unroll
        for (int k = 0; k < 3; ++k) {
            int j = tid + TPB * k;           // 0..767, region boundaries wave-uniform
            uint64_t g;
            uint32_t l;
            if (j < 448)      { g = em_g + (uint64_t)j * 16;          l = em_l + j * 16; }
            else if (j < 640) { int q = j - 448; g = fa_g + (uint64_t)q * 16; l = fa_l + q * 16; }
            else if (j < 704) { int q = j - 640; g = ft_g + (uint64_t)q * 16; l = ft_l + q * 16; }
            else              { int q = j - 704; g = et_g + (uint64_t)q * 16; l = et_l + q * 16; }
            async_b128(l, g);                 // exactly 3 async ops per wave per tile
        }
    };

    float acc = 0.0f;
    int tile = blockIdx.x;
    int buf = 0;
    if (tile < ntiles) issue_tile(tile, 0);

    for (; tile < ntiles; tile += gridDim.x) {
        int nt = tile + (int)gridDim.x;
        bool pf = (nt < ntiles);
        if (pf) { issue_tile(nt, buf ^ 1); wait_async3(); }
        else    { wait_async0(); }
        __syncthreads();                     // all waves' tile data resident in LDS

        const float* em = &sm.em[buf][tid * 7];   // stride 7: conflict-free (gcd(7,64)=1)
        const float* fa = &sm.fa[buf][tid * 3];   // stride 3: conflict-free
        int ft = sm.ft[buf][tid];
        int et = sm.et[buf][tid];
        acc += row_loss(fa[0], fa[1], fa[2], em, ft, et, sm);

        __syncthreads();                     // done reading buf before it is refilled
        buf ^= 1;
    }

    __syncthreads();                         // td/ent visible even when ntiles==0

    // Tail rows (B % 256), direct loads, block 0 only (B = 2^21 -> empty)
    if (blockIdx.x == 0) {
        for (int r = ntiles * ROWS + tid; r < B; r += TPB) {
            float f0 = fat[r * 3 + 0], f1 = fat[r * 3 + 1], f2 = fat[r * 3 + 2];
            float e[7];
#pragma unroll
            for (int j = 0; j < 7; ++j) e[j] = emo[r * 7 + j];
            acc += row_loss(f0, f1, f2, e, ftg[r], etg[r], sm);
        }
    }

    // Fixed-order block reduction -> one double partial per block (deterministic)
    sm.sred[tid] = acc;
    __syncthreads();
#pragma unroll
    for (int s = TPB / 2; s > 0; s >>= 1) {
        if (tid < s) sm.sred[tid] += sm.sred[tid + s];
        __syncthreads();
    }
    if (tid == 0) part[blockIdx.x] = (double)sm.sred[0];
}

__global__ void __launch_bounds__(TPB)
mt_loss_final(const double* __restrict__ part, int n, float* __restrict__ out, int B) {
    __shared__ double sd[TPB];
    double s = 0.0;
    for (int i = threadIdx.x; i < n; i += TPB) s += part[i];
    sd[threadIdx.x] = s;
    __syncthreads();
#pragma unroll
    for (int k = TPB / 2; k > 0; k >>= 1) {
        if (threadIdx.x < k) sd[threadIdx.x] += sd[threadIdx.x + k];
        __syncthreads();
    }
    if (threadIdx.x == 0) out[0] = (float)(sd[0] / (double)B);
}

extern "C" void kernel_launch(void* const* d_in, const int* in_sizes, int n_in,
                              void* d_out, int out_size, void* d_ws, size_t ws_size,
                              hipStream_t stream) {
    const float* fat = (const float*)d_in[0];   // [B,3] fp32
    const float* emo = (const float*)d_in[1];   // [B,7] fp32
    const int*   ftg = (const int*)d_in[2];     // [B] i32
    const int*   etg = (const int*)d_in[3];     // [B] i32
    int B = in_sizes[0] / 3;

    int ntiles = B / ROWS;
    int nblk = 2048;
    int maxblk = (int)(ws_size / sizeof(double));
    if (nblk > maxblk) nblk = maxblk;
    if (ntiles > 0 && nblk > ntiles) nblk = ntiles;
    if (nblk < 1) nblk = 1;

    double* part = (double*)d_ws;
    mt_loss_main<<<nblk, TPB, 0, stream>>>(fat, emo, ftg, etg, B, part);
    mt_loss_final<<<1, TPB, 0, stream>>>(part, nblk, (float*)d_out, B);
}